// AttentionSE3_51110110822703
// MI455X (gfx1250) — compile-verified
//
#include <hip/hip_runtime.h>
#include <math.h>

typedef __attribute__((ext_vector_type(2))) float v2f;
typedef __attribute__((ext_vector_type(8))) float v8f;

#define HEADS 8
#define DKIM  16
#define FEAT  128   // HEADS*DKIM == CV*DV
#define CV    32
#define DV    4

// ---- rowptr[n] = lower_bound(edge_dst, n); edges are sorted by dst ----
__global__ void rowptr_kernel(const int* __restrict__ edge_dst, int* __restrict__ rowptr,
                              int N, int E) {
  int n = blockIdx.x * blockDim.x + threadIdx.x;
  if (n > N) return;
  int lo = 0, hi = E;
  while (lo < hi) { int mid = (lo + hi) >> 1; if (edge_dst[mid] < n) lo = mid + 1; else hi = mid; }
  rowptr[n] = lo;
}

// ---- per-edge, per-head dot(key, q[dst]) via V_WMMA_F32_16X16X4_F32 ----
// One wave handles 16 edges x 8 heads. A = key tile (16 edges x K16),
// B = fused-query tile transposed (K16 x 16 edges); diagonal of D = desired dots.
__global__ void __launch_bounds__(32)
ew_kernel(const float* __restrict__ key, const float* __restrict__ q0,
          const float* __restrict__ q1, const int* __restrict__ edge_dst,
          float* __restrict__ ew, int E) {
  __shared__ __align__(16) float sKey[16 * FEAT];  // 8 KB
  __shared__ __align__(16) float sQ[16 * FEAT];    // 8 KB
  __shared__ float sD[8 * 32];                     // diagonal bounce
  const int lane = threadIdx.x;
  const int e0 = blockIdx.x * 16;

  // Stage: coalesced b128 key rows; fused query gather (concat(q0,q1) reshape):
  // flat feature f = c*4 + r; lane == channel c, so lane owns f = 4*lane..4*lane+3.
  for (int r = 0; r < 16; ++r) {
    int er = e0 + r; if (er >= E) er = E - 1;
    ((float4*)sKey)[r * 32 + lane] = ((const float4*)(key + (size_t)er * FEAT))[lane];
    int n = edge_dst[er];
    sQ[r * FEAT + lane * 4 + 0] = q0[(size_t)n * CV + lane];
    const float* q1p = q1 + ((size_t)n * CV + lane) * 3;
    sQ[r * FEAT + lane * 4 + 1] = q1p[0];
    sQ[r * FEAT + lane * 4 + 2] = q1p[1];
    sQ[r * FEAT + lane * 4 + 3] = q1p[2];
  }
  __syncthreads();

  // ISA 7.12.2: 32-bit A 16x4 -> lane l holds row (l&15), K pair (l<16 ? {0,1} : {2,3});
  // B 4x16 mirrors (row-in-VGPR striping), so both operands use the same fetch pattern.
  const int m = lane & 15;
  const int koff = (lane < 16) ? 0 : 2;
  const float scale = 0.08838834764831845f;  // 1/sqrt(KEY_FEATS=128)

  for (int h = 0; h < HEADS; ++h) {
    v8f c = {0.f, 0.f, 0.f, 0.f, 0.f, 0.f, 0.f, 0.f};
#pragma unroll
    for (int kk = 0; kk < 4; ++kk) {
      const int col = h * DKIM + kk * 4 + koff;
      v2f a, b;
      a.x = sKey[m * FEAT + col]; a.y = sKey[m * FEAT + col + 1];
      b.x = sQ[m * FEAT + col];   b.y = sQ[m * FEAT + col + 1];
      c = __builtin_amdgcn_wmma_f32_16x16x4_f32(false, a, false, b, (short)0, c,
                                                false, false);
    }
#pragma unroll
    for (int g = 0; g < 8; ++g) sD[g * 32 + lane] = c[g];
    __syncthreads();
    // D layout: VGPR g, lanes 0-15 -> (M=g, N=lane); lanes 16-31 -> (M=g+8, N=lane-16)
    if (lane < 16 && (e0 + lane) < E) {
      float d = (lane < 8) ? sD[lane * 32 + lane] : sD[(lane - 8) * 32 + lane + 16];
      ew[(size_t)(e0 + lane) * HEADS + h] = d * scale;
    }
    __syncthreads();
  }
}

// ---- numerically-stable segmented softmax stats: one thread per (node, head) ----
__global__ void stats_kernel(const float* __restrict__ ew, const int* __restrict__ rowptr,
                             float* __restrict__ mbuf, float* __restrict__ dbuf, int NH) {
  int t = blockIdx.x * blockDim.x + threadIdx.x;
  if (t >= NH) return;
  int n = t >> 3, h = t & 7;
  int s = rowptr[n], e = rowptr[n + 1];
  float mx = -INFINITY, den = 0.f;
  for (int i = s; i < e; ++i) {
    float x = ew[(size_t)i * HEADS + h];
    if (x > mx) { den = den * __expf(mx - x) + 1.f; mx = x; }
    else        { den += __expf(x - mx); }
  }
  mbuf[t] = mx;
  dbuf[t] = den;
}

// ---- weighted segment-sum of values: one wave per node, lane == channel ----
__global__ void __launch_bounds__(256)
out_kernel(const float* __restrict__ value, const float* __restrict__ ew,
           const int* __restrict__ rowptr, const float* __restrict__ mbuf,
           const float* __restrict__ dbuf, float* __restrict__ out, int N) {
  int wave = threadIdx.x >> 5, lane = threadIdx.x & 31;
  int n = blockIdx.x * 8 + wave;
  if (n >= N) return;
  int s = rowptr[n], e = rowptr[n + 1];
  int h = lane >> 2;                       // head for channel `lane` (c*DV+d in [16h,16h+16))
  float mh = mbuf[n * HEADS + h];
  float dh = dbuf[n * HEADS + h];
  float rd = (dh > 0.f) ? 1.f / dh : 0.f;
  float4 acc = make_float4(0.f, 0.f, 0.f, 0.f);
  for (int i = s; i < e; ++i) {
    float w = __expf(ew[(size_t)i * HEADS + h] - mh) * rd;
    float4 v4 = ((const float4*)value)[(size_t)i * 32 + lane];  // coalesced 512B/edge
    acc.x += w * v4.x; acc.y += w * v4.y; acc.z += w * v4.z; acc.w += w * v4.w;
  }
  // unfuse: feat[...,:1] then feat[...,1:] concatenated flat in d_out
  out[(size_t)n * CV + lane] = acc.x;
  float* o1 = out + (size_t)N * CV + ((size_t)n * CV + lane) * 3;
  o1[0] = acc.y; o1[1] = acc.z; o1[2] = acc.w;
}

extern "C" void kernel_launch(void* const* d_in, const int* in_sizes, int n_in,
                              void* d_out, int out_size, void* d_ws, size_t ws_size,
                              hipStream_t stream) {
  const float* key = (const float*)d_in[0];
  const float* q0  = (const float*)d_in[1];
  const float* q1  = (const float*)d_in[2];
  const float* val = (const float*)d_in[3];
  const int* edge_dst = (const int*)d_in[4];
  const int E = in_sizes[4];
  const int N = in_sizes[1] / CV;

  // workspace: ew (E*8 f32) | m (N*8 f32) | den (N*8 f32) | rowptr (N+1 i32)  ~22 MB
  float* ew   = (float*)d_ws;
  float* mbuf = ew + (size_t)E * HEADS;
  float* dbuf = mbuf + (size_t)N * HEADS;
  int*  rowptr = (int*)(dbuf + (size_t)N * HEADS);

  rowptr_kernel<<<(N + 1 + 255) / 256, 256, 0, stream>>>(edge_dst, rowptr, N, E);
  ew_kernel<<<(E + 15) / 16, 32, 0, stream>>>(key, q0, q1, edge_dst, ew, E);
  stats_kernel<<<(N * HEADS + 255) / 256, 256, 0, stream>>>(ew, rowptr, mbuf, dbuf, N * HEADS);
  out_kernel<<<(N + 7) / 8, 256, 0, stream>>>(val, ew, rowptr, mbuf, dbuf, (float*)d_out, N);
}